// GAT_60936995995661
// MI455X (gfx1250) — compile-verified
//
#include <hip/hip_runtime.h>
#include <hip/hip_bf16.h>
#include <math.h>

typedef __attribute__((ext_vector_type(2))) float v2f;
typedef __attribute__((ext_vector_type(8))) float v8f;

#define NEG_SLOPE 0.2f

// ---------------- utility ----------------

__device__ __forceinline__ void atomicMaxF32(float* addr, float val) {
    // correct for all finite floats if *addr initialized to -inf bits (0xFF800000)
    if (val >= 0.0f) atomicMax((int*)addr, __float_as_int(val));
    else             atomicMin((unsigned int*)addr, __float_as_uint(val));
}

__global__ void gat_fill_f32(float* __restrict__ p, long n, float v) {
    long i = (long)blockIdx.x * blockDim.x + threadIdx.x;
    if (i < n) p[i] = v;
}

// Compact int64 edge list -> int32, materializing self-loops (removes branch +
// halves index bytes for the 6 hot edge passes).
__global__ void gat_prep_edges(const long long* __restrict__ ei, int E, int Etot,
                               int* __restrict__ src32, int* __restrict__ dst32) {
    int i = blockIdx.x * blockDim.x + threadIdx.x;
    if (i >= Etot) return;
    if (i < E) { src32[i] = (int)ei[i]; dst32[i] = (int)ei[(size_t)E + i]; }
    else       { src32[i] = dst32[i] = i - E; }
}

// ---------------- WMMA GEMMs (f32, 16x16x4) ----------------
// Layer 1: h1[N,64] = x[N,128] @ W1[128,64].  4 waves/block, one 16x16 tile per wave.
__global__ __launch_bounds__(128)
void gat_gemm1(const float* __restrict__ x, const float* __restrict__ W,
               float* __restrict__ h, int N) {
    const int lane = threadIdx.x & 31;
    const int wave = threadIdx.x >> 5;
    const int hs   = lane >> 4;       // half-select: 0 -> K pair {0,1}, 1 -> {2,3}
    const int m    = lane & 15;
    const int col0 = wave * 16;       // 4 waves cover 64 cols
    const int row0 = blockIdx.x * 16;
    int arow = row0 + m; if (arow >= N) arow = N - 1;   // clamp (no EXEC change)
    const float* __restrict__ xr = x + (size_t)arow * 128;

    v8f c = {};
#pragma unroll
    for (int kb = 0; kb < 128; kb += 4) {
        const int k0 = kb + 2 * hs;
        v2f a, b;
        a[0] = xr[k0];
        a[1] = xr[k0 + 1];
        b[0] = W[(size_t)k0       * 64 + col0 + m];
        b[1] = W[(size_t)(k0 + 1) * 64 + col0 + m];
        c = __builtin_amdgcn_wmma_f32_16x16x4_f32(false, a, false, b,
                                                  (short)0, c, false, false);
    }
#pragma unroll
    for (int r = 0; r < 8; ++r) {
        const int row = row0 + r + 8 * hs;
        if (row < N) h[(size_t)row * 64 + col0 + m] = c[r];
    }
}

// Layer 2: h2[N,40] = xh[N,64] @ W2[64,40].  3 waves/block cover 48 cols (store <40).
__global__ __launch_bounds__(96)
void gat_gemm2(const float* __restrict__ xh, const float* __restrict__ W,
               float* __restrict__ h2, int N) {
    const int lane = threadIdx.x & 31;
    const int wave = threadIdx.x >> 5;
    const int hs   = lane >> 4;
    const int m    = lane & 15;
    const int col0 = wave * 16;
    const int col  = col0 + m;
    const int row0 = blockIdx.x * 16;
    int arow = row0 + m; if (arow >= N) arow = N - 1;
    const float* __restrict__ xr = xh + (size_t)arow * 64;

    v8f c = {};
#pragma unroll
    for (int kb = 0; kb < 64; kb += 4) {
        const int k0 = kb + 2 * hs;
        v2f a, b;
        a[0] = xr[k0];
        a[1] = xr[k0 + 1];
        b[0] = (col < 40) ? W[(size_t)k0       * 40 + col] : 0.0f;
        b[1] = (col < 40) ? W[(size_t)(k0 + 1) * 40 + col] : 0.0f;
        c = __builtin_amdgcn_wmma_f32_16x16x4_f32(false, a, false, b,
                                                  (short)0, c, false, false);
    }
    if (col < 40) {
#pragma unroll
        for (int r = 0; r < 8; ++r) {
            const int row = row0 + r + 8 * hs;
            if (row < N) h2[(size_t)row * 40 + col] = c[r];
        }
    }
}

// ---------------- per-node attention logit projections ----------------
// as[n*H+h] = dot(h[n,h,:], a_src[h,:]) ; ad likewise.  One thread per (node,head).
template <int H, int C>
__global__ void gat_alpha(const float* __restrict__ h,
                          const float* __restrict__ a_s, const float* __restrict__ a_d,
                          float* __restrict__ as, float* __restrict__ ad, int N) {
    long t = (long)blockIdx.x * blockDim.x + threadIdx.x;
    if (t >= (long)N * H) return;
    const int n = (int)(t / H), hh = (int)(t % H);
    const float* __restrict__ hp = h + (size_t)n * H * C + (size_t)hh * C;
    float s = 0.0f, d = 0.0f;
#pragma unroll
    for (int c = 0; c < C; ++c) {
        const float v = hp[c];
        s += v * a_s[hh * C + c];
        d += v * a_d[hh * C + c];
    }
    as[t] = s; ad[t] = d;
}

// ---------------- edge pass 1: segment max ----------------
template <int H>
__global__ void gat_edge_max(const int* __restrict__ src, const int* __restrict__ dst, int Etot,
                             const float* __restrict__ as, const float* __restrict__ ad,
                             float* __restrict__ emax) {
    const long total = (long)Etot * H;
    for (long t = (long)blockIdx.x * blockDim.x + threadIdx.x; t < total;
         t += (long)gridDim.x * blockDim.x) {
        const long e = t / H; const int hh = (int)(t % H);       // H pow2 / 1 -> shifts
        const long s = src[e], d = dst[e];
        float v = as[s * H + hh] + ad[d * H + hh];
        v = (v > 0.0f) ? v : NEG_SLOPE * v;
        atomicMaxF32(&emax[d * H + hh], v);
    }
}

// ---------------- edge pass 2: softmax denominator ----------------
template <int H>
__global__ void gat_edge_sum(const int* __restrict__ src, const int* __restrict__ dst, int Etot,
                             const float* __restrict__ as, const float* __restrict__ ad,
                             const float* __restrict__ emax, float* __restrict__ den) {
    const long total = (long)Etot * H;
    for (long t = (long)blockIdx.x * blockDim.x + threadIdx.x; t < total;
         t += (long)gridDim.x * blockDim.x) {
        const long e = t / H; const int hh = (int)(t % H);
        const long s = src[e], d = dst[e];
        float v = as[s * H + hh] + ad[d * H + hh];
        v = (v > 0.0f) ? v : NEG_SLOPE * v;
        atomicAdd(&den[d * H + hh], __expf(v - emax[d * H + hh]));
    }
}

// ---------------- edge pass 3: alpha * h[src] scatter-add (one wave32 per edge) ----------------
template <int H, int C>
__global__ void gat_edge_agg(const int* __restrict__ src, const int* __restrict__ dst, int Etot,
                             const float* __restrict__ as, const float* __restrict__ ad,
                             const float* __restrict__ emax, const float* __restrict__ den,
                             const float* __restrict__ h, float* __restrict__ out) {
    constexpr int HC = H * C;
    const int lane = threadIdx.x & 31;
    const long wid = ((long)blockIdx.x * blockDim.x + threadIdx.x) >> 5;
    const long nw  = ((long)gridDim.x * blockDim.x) >> 5;
    for (long e = wid; e < Etot; e += nw) {
        const long s = src[e], d = dst[e];
#pragma unroll
        for (int jj = 0; jj < (HC + 31) / 32; ++jj) {
            const int j = lane + 32 * jj;
            if (j < HC) {
                const int hh = j / C;                             // compile-time C
                float v = as[s * H + hh] + ad[d * H + hh];
                v = (v > 0.0f) ? v : NEG_SLOPE * v;
                const float alpha = __expf(v - emax[d * H + hh]) / (den[d * H + hh] + 1e-16f);
                atomicAdd(&out[d * (size_t)HC + j], h[s * (size_t)HC + j] * alpha);
            }
        }
    }
}

// ---------------- epilogues ----------------
template <int HC>
__global__ void gat_relu_bias(float* __restrict__ p, const float* __restrict__ b, long n) {
    long i = (long)blockIdx.x * blockDim.x + threadIdx.x;
    if (i < n) p[i] = fmaxf(p[i] + b[i % HC], 0.0f);
}
template <int HC>
__global__ void gat_bias(float* __restrict__ p, const float* __restrict__ b, long n) {
    long i = (long)blockIdx.x * blockDim.x + threadIdx.x;
    if (i < n) p[i] = p[i] + b[i % HC];
}

// ---------------- launcher ----------------
extern "C" void kernel_launch(void* const* d_in, const int* in_sizes, int n_in,
                              void* d_out, int out_size, void* d_ws, size_t ws_size,
                              hipStream_t stream) {
    const float*     x      = (const float*)d_in[0];
    const long long* ei     = (const long long*)d_in[1];
    const float*     W1     = (const float*)d_in[2];
    const float*     a_src1 = (const float*)d_in[3];
    const float*     a_dst1 = (const float*)d_in[4];
    const float*     b1     = (const float*)d_in[5];
    const float*     W2     = (const float*)d_in[6];
    const float*     a_src2 = (const float*)d_in[7];
    const float*     a_dst2 = (const float*)d_in[8];
    const float*     b2     = (const float*)d_in[9];

    const int N    = in_sizes[0] / 128;   // 100000
    const int E    = in_sizes[1] / 2;     // 1600000
    const int Etot = E + N;               // + self loops

    // workspace carve-up
    char* ws = (char*)d_ws;
    float* h1    = (float*)ws; ws += (size_t)N * 64 * sizeof(float);
    float* as1   = (float*)ws; ws += (size_t)N * 8  * sizeof(float);
    float* ad1   = (float*)ws; ws += (size_t)N * 8  * sizeof(float);
    float* emax1 = (float*)ws; ws += (size_t)N * 8  * sizeof(float);
    float* den1  = (float*)ws; ws += (size_t)N * 8  * sizeof(float);
    float* xh    = (float*)ws; ws += (size_t)N * 64 * sizeof(float);  // layer-1 agg + relu
    float* h2    = (float*)ws; ws += (size_t)N * 40 * sizeof(float);
    float* as2   = (float*)ws; ws += (size_t)N * sizeof(float);
    float* ad2   = (float*)ws; ws += (size_t)N * sizeof(float);
    float* emax2 = (float*)ws; ws += (size_t)N * sizeof(float);
    float* den2  = (float*)ws; ws += (size_t)N * sizeof(float);
    int*   src32 = (int*)ws;   ws += (size_t)Etot * sizeof(int);
    int*   dst32 = (int*)ws;   ws += (size_t)Etot * sizeof(int);
    float* out   = (float*)d_out;

    const int TB = 256;
    auto blocks = [](long n, int tb) { return (int)((n + tb - 1) / tb); };

    // init + edge compaction
    gat_prep_edges<<<blocks(Etot, TB), TB, 0, stream>>>(ei, E, Etot, src32, dst32);
    gat_fill_f32<<<blocks((long)N * 64, TB), TB, 0, stream>>>(xh,    (long)N * 64, 0.0f);
    gat_fill_f32<<<blocks((long)N * 8,  TB), TB, 0, stream>>>(emax1, (long)N * 8,  -INFINITY);
    gat_fill_f32<<<blocks((long)N * 8,  TB), TB, 0, stream>>>(den1,  (long)N * 8,  0.0f);
    gat_fill_f32<<<blocks((long)N,      TB), TB, 0, stream>>>(emax2, (long)N,      -INFINITY);
    gat_fill_f32<<<blocks((long)N,      TB), TB, 0, stream>>>(den2,  (long)N,      0.0f);
    gat_fill_f32<<<blocks((long)N * 40, TB), TB, 0, stream>>>(out,   (long)N * 40, 0.0f);

    // ---- layer 1 (H=8, C=8) ----
    gat_gemm1<<<(N + 15) / 16, 128, 0, stream>>>(x, W1, h1, N);
    gat_alpha<8, 8><<<blocks((long)N * 8, TB), TB, 0, stream>>>(h1, a_src1, a_dst1, as1, ad1, N);
    gat_edge_max<8><<<blocks((long)Etot * 8, TB), TB, 0, stream>>>(src32, dst32, Etot, as1, ad1, emax1);
    gat_edge_sum<8><<<blocks((long)Etot * 8, TB), TB, 0, stream>>>(src32, dst32, Etot, as1, ad1, emax1, den1);
    gat_edge_agg<8, 8><<<blocks((long)Etot * 32, TB), TB, 0, stream>>>(src32, dst32, Etot,
                                                                       as1, ad1, emax1, den1, h1, xh);
    gat_relu_bias<64><<<blocks((long)N * 64, TB), TB, 0, stream>>>(xh, b1, (long)N * 64);

    // ---- layer 2 (H=1, C=40) ----
    gat_gemm2<<<(N + 15) / 16, 96, 0, stream>>>(xh, W2, h2, N);
    gat_alpha<1, 40><<<blocks((long)N, TB), TB, 0, stream>>>(h2, a_src2, a_dst2, as2, ad2, N);
    gat_edge_max<1><<<blocks((long)Etot, TB), TB, 0, stream>>>(src32, dst32, Etot, as2, ad2, emax2);
    gat_edge_sum<1><<<blocks((long)Etot, TB), TB, 0, stream>>>(src32, dst32, Etot, as2, ad2, emax2, den2);
    gat_edge_agg<1, 40><<<blocks((long)Etot * 32, TB), TB, 0, stream>>>(src32, dst32, Etot,
                                                                        as2, ad2, emax2, den2, h2, out);
    gat_bias<40><<<blocks((long)N * 40, TB), TB, 0, stream>>>(out, b2, (long)N * 40);
}